// SSAttention_46548855554662
// MI455X (gfx1250) — compile-verified
//
#include <hip/hip_runtime.h>
#include <math.h>

// Problem constants (B=1)
#define C_    64
#define Hh    96
#define Ww    96
#define HW    9216
#define BETA  0.3f

// Flash-attention tiling
#define WAVES 4                    // waves per block (128 threads)
#define SPLIT 4                    // split-K over the column sweep
#define MBLK  (HW / (16 * WAVES))  // 144 M-blocks
#define NITER (HW / SPLIT / 64)    // 36 column-tiles of 64 per split
#define PITCH 68                   // LDS pitch (floats): 272B rows -> 16B-aligned chunks,
                                   // conflict-free b64 fragment reads (banks 4n+4k+2h)

typedef __attribute__((ext_vector_type(2))) float v2f;
typedef __attribute__((ext_vector_type(8))) float v8f;

// ---------------------------------------------------------------------------
// CDNA5 async global->LDS copy (ASYNCcnt-tracked), 16B per lane.
// ---------------------------------------------------------------------------
__device__ __forceinline__ void async_copy16(unsigned lds_addr, const void* gptr)
{
    asm volatile("global_load_async_to_lds_b128 %0, %1, off"
                 :
                 : "v"(lds_addr), "v"((unsigned long long)(uintptr_t)gptr)
                 : "memory");
}
__device__ __forceinline__ void wait_async()
{
    asm volatile("s_wait_asynccnt 0" ::: "memory");
}

// ---------------------------------------------------------------------------
// Kernel 1: dilated 3x3 conv (d=2, p=2) -> y ; 1x1 conv -> v.
// Emits y natural (C,HW) for Q-view reads, plus pre-transposed copies:
//   yT[n*64 + c] = yflat[c*HW + n]   (K-view rows, contiguous per n)
//   vT[c*HW + n] = vflat[n*64 + c]   (V-view cols, contiguous per c)
// so the flash kernel can stage tiles with contiguous 16B async copies.
// ---------------------------------------------------------------------------
__global__ __launch_bounds__(256) void ssa_conv_kernel(
    const float* __restrict__ x,  const float* __restrict__ dW,
    const float* __restrict__ db, const float* __restrict__ cW,
    const float* __restrict__ cb, float* __restrict__ y, float* __restrict__ v,
    float* __restrict__ yT, float* __restrict__ vT)
{
    __shared__ __align__(16) float sdW[C_ * 9];
    __shared__ __align__(16) float scW[C_];
    const int tid  = threadIdx.x;
    const int oc   = blockIdx.x / (HW / 256);
    const int tile = blockIdx.x % (HW / 256);

    for (int i = tid; i < C_ * 9; i += 256) sdW[i] = dW[oc * C_ * 9 + i];
    if (tid < C_) scW[tid] = cW[oc * C_ + tid];
    __syncthreads();

    const int pos = tile * 256 + tid;
    const int h = pos / Ww, w = pos % Ww;
    float acc  = db[oc];
    float vacc = cb[oc];
    for (int ic = 0; ic < C_; ++ic) {
        const float* xp = x + ic * HW;
        vacc = fmaf(xp[pos], scW[ic], vacc);
        const float* wk = &sdW[ic * 9];
        #pragma unroll
        for (int kh = 0; kh < 3; ++kh) {
            const int ih = h + 2 * kh - 2;
            if (ih < 0 || ih >= Hh) continue;
            #pragma unroll
            for (int kw = 0; kw < 3; ++kw) {
                const int iw = w + 2 * kw - 2;
                if (iw < 0 || iw >= Ww) continue;
                acc = fmaf(xp[ih * Ww + iw], wk[kh * 3 + kw], acc);
            }
        }
    }
    y[oc * HW + pos]            = acc;
    yT[(size_t)pos * C_ + oc]   = acc;   // K-view transpose
    v[oc * HW + pos]            = vacc;
    // vflat index i = oc*HW+pos ; V-view: row i/64, col i%64 -> vT[col*HW + row]
    vT[(size_t)(pos & 63) * HW + oc * (HW / 64) + (pos >> 6)] = vacc;
}

// ---------------------------------------------------------------------------
// Kernel 2: fused flash attention, fp32 WMMA 16x16x4, double-buffered
// async-to-LDS K/V staging. Each wave owns a 16-row M tile.
// ---------------------------------------------------------------------------
__global__ __launch_bounds__(WAVES * 32) void ssa_flash_kernel(
    const float* __restrict__ y, const float* __restrict__ yT,
    const float* __restrict__ vT,
    float* __restrict__ Op, float* __restrict__ mp, float* __restrict__ lp)
{
    __shared__ __align__(16) float sK[2][64 * PITCH];        // sK[b][n][c]
    __shared__ __align__(16) float sV[2][64 * PITCH];        // sV[b][c][n]
    __shared__ __align__(16) float sP[WAVES * 16 * PITCH];   // per-wave P tile [m][n]

    const int tid   = threadIdx.x;
    const int wave  = tid >> 5;
    const int lane  = tid & 31;
    const int half  = (lane < 16) ? 0 : 1;   // A/B layout half selector
    const int l16   = lane & 15;
    const int split = blockIdx.x / MBLK;
    const int mblk  = blockIdx.x % MBLK;
    const int m0    = (mblk * WAVES + wave) * 16;
    const int nbase = split * (HW / SPLIT);
    float* sPw = sP + wave * 16 * PITCH;

    // Issue async copies for one 64-column K/V tile into buffer `buf`.
    auto prefetch_tile = [&](int buf, int n0) {
        #pragma unroll
        for (int j = 0; j < 8; ++j) {                 // 1024 chunks / 128 threads
            const int q = tid + j * (WAVES * 32);
            const int n = q >> 4, c4 = (q & 15) * 4;
            async_copy16((unsigned)(uintptr_t)&sK[buf][n * PITCH + c4],
                         yT + (size_t)(n0 + n) * C_ + c4);
        }
        #pragma unroll
        for (int j = 0; j < 8; ++j) {
            const int q = tid + j * (WAVES * 32);
            const int c = q >> 4, n4 = (q & 15) * 4;
            async_copy16((unsigned)(uintptr_t)&sV[buf][c * PITCH + n4],
                         vT + (size_t)c * HW + n0 + n4);
        }
    };

    // Q tile (16x64) in A-layout register fragments: Q[n][c] = yflat[n*64 + c]
    v2f aQ[16];
    {
        const int m = m0 + l16;
        #pragma unroll
        for (int kc = 0; kc < 16; ++kc)
            aQ[kc] = *(const v2f*)(y + (size_t)m * C_ + 4 * kc + 2 * half);
    }

    v8f o[4];
    #pragma unroll
    for (int ct = 0; ct < 4; ++ct)
        #pragma unroll
        for (int e = 0; e < 8; ++e) o[ct][e] = 0.0f;
    float m_i[8], l_i[8];
    #pragma unroll
    for (int j = 0; j < 8; ++j) { m_i[j] = -INFINITY; l_i[j] = 0.0f; }

    int cur = 0;
    prefetch_tile(0, nbase);

    for (int it = 0; it < NITER; ++it) {
        wait_async();            // own async copies into sK/sV[cur] landed
        __syncthreads();         // every wave's copies landed; prev compute done
        if (it + 1 < NITER)      // overlap next tile's DMA with this compute
            prefetch_tile(cur ^ 1, nbase + (it + 1) * 64);

        const float* __restrict__ Kt = sK[cur];
        const float* __restrict__ Vt = sV[cur];

        // S = Q @ K_tile (16 x 64): 4 col sub-tiles, K-dim 64 in chunks of 4
        v8f s[4];
        #pragma unroll
        for (int nt = 0; nt < 4; ++nt) {
            v8f acc;
            #pragma unroll
            for (int e = 0; e < 8; ++e) acc[e] = 0.0f;
            const int n = nt * 16 + l16;
            #pragma unroll
            for (int kc = 0; kc < 16; ++kc) {
                const v2f b = *(const v2f*)&Kt[n * PITCH + 4 * kc + 2 * half];
                acc = __builtin_amdgcn_wmma_f32_16x16x4_f32(
                        false, aQ[kc], false, b, (short)0, acc, false, false);
            }
            s[nt] = acc;
        }

        // Threshold mask (|s|>BETA ? s : 0) + online softmax update.
        // C/D layout: VGPR j holds row (j + 8*half), col = l16 across 16 lanes.
        #pragma unroll
        for (int j = 0; j < 8; ++j) {
            float mx = -INFINITY;
            #pragma unroll
            for (int nt = 0; nt < 4; ++nt) {
                float val = s[nt][j];
                val = (fabsf(val) > BETA) ? val : 0.0f;
                s[nt][j] = val;
                mx = fmaxf(mx, val);
            }
            mx = fmaxf(mx, __shfl_xor(mx, 1));
            mx = fmaxf(mx, __shfl_xor(mx, 2));
            mx = fmaxf(mx, __shfl_xor(mx, 4));
            mx = fmaxf(mx, __shfl_xor(mx, 8));

            const float mnew  = fmaxf(m_i[j], mx);
            const float scale = __expf(m_i[j] - mnew);
            m_i[j] = mnew;
            l_i[j] *= scale;
            #pragma unroll
            for (int ct = 0; ct < 4; ++ct) o[ct][j] *= scale;

            float rs = 0.0f;
            #pragma unroll
            for (int nt = 0; nt < 4; ++nt) {
                const float pv = __expf(s[nt][j] - mnew);
                s[nt][j] = pv;
                rs += pv;
            }
            rs += __shfl_xor(rs, 1);
            rs += __shfl_xor(rs, 2);
            rs += __shfl_xor(rs, 4);
            rs += __shfl_xor(rs, 8);
            l_i[j] += rs;

            // Spill P row to wave-private LDS for C/D -> A relayout
            const int m = j + half * 8;
            #pragma unroll
            for (int nt = 0; nt < 4; ++nt)
                sPw[m * PITCH + nt * 16 + l16] = s[nt][j];
        }

        // O += P @ V_tile (16 x 64): K-dim 64 (tile cols) in chunks of 4
        #pragma unroll
        for (int ct = 0; ct < 4; ++ct) {
            v8f acc = o[ct];
            #pragma unroll
            for (int kc = 0; kc < 16; ++kc) {
                const v2f a = *(const v2f*)&sPw[l16 * PITCH + 4 * kc + 2 * half];
                const v2f b = *(const v2f*)&Vt[(ct * 16 + l16) * PITCH + 4 * kc + 2 * half];
                acc = __builtin_amdgcn_wmma_f32_16x16x4_f32(
                        false, a, false, b, (short)0, acc, false, false);
            }
            o[ct] = acc;
        }
        cur ^= 1;
    }

    // Epilogue: write partials. Op layout [split][c][n] for coalesced combine.
    #pragma unroll
    for (int j = 0; j < 8; ++j) {
        const int row = m0 + j + half * 8;
        #pragma unroll
        for (int ct = 0; ct < 4; ++ct) {
            const int c = ct * 16 + l16;
            Op[(size_t)(split * C_ + c) * HW + row] = o[ct][j];
        }
        if (l16 == 0) {
            mp[split * HW + row] = m_i[j];
            lp[split * HW + row] = l_i[j];
        }
    }
}

// ---------------------------------------------------------------------------
// Kernel 3: merge split-K partials (log-sum-exp combine) + residual add.
// ---------------------------------------------------------------------------
__global__ __launch_bounds__(256) void ssa_combine_kernel(
    const float* __restrict__ x, const float* __restrict__ Op,
    const float* __restrict__ mp, const float* __restrict__ lp,
    float* __restrict__ out)
{
    const int gid = blockIdx.x * 256 + threadIdx.x;
    const int n = gid % HW;
    const int c = gid / HW;
    float mloc[SPLIT];
    float M = -INFINITY;
    #pragma unroll
    for (int s = 0; s < SPLIT; ++s) {
        mloc[s] = mp[s * HW + n];
        M = fmaxf(M, mloc[s]);
    }
    float L = 0.0f, acc = 0.0f;
    #pragma unroll
    for (int s = 0; s < SPLIT; ++s) {
        const float w = __expf(mloc[s] - M);
        L   = fmaf(lp[s * HW + n], w, L);
        acc = fmaf(Op[(size_t)(s * C_ + c) * HW + n], w, acc);
    }
    out[gid] = x[gid] + acc / L;
}

// ---------------------------------------------------------------------------
extern "C" void kernel_launch(void* const* d_in, const int* in_sizes, int n_in,
                              void* d_out, int out_size, void* d_ws, size_t ws_size,
                              hipStream_t stream)
{
    (void)in_sizes; (void)n_in; (void)out_size; (void)ws_size;
    const float* x  = (const float*)d_in[0];
    const float* dW = (const float*)d_in[1];
    const float* db = (const float*)d_in[2];
    const float* cW = (const float*)d_in[3];
    const float* cb = (const float*)d_in[4];

    float* ws = (float*)d_ws;
    float* y  = ws;                               // C_*HW
    float* v  = y  + (size_t)C_ * HW;             // C_*HW
    float* yT = v  + (size_t)C_ * HW;             // C_*HW (K-view transpose)
    float* vT = yT + (size_t)C_ * HW;             // C_*HW (V-view transpose)
    float* Op = vT + (size_t)C_ * HW;             // SPLIT*C_*HW
    float* mp = Op + (size_t)SPLIT * C_ * HW;     // SPLIT*HW
    float* lp = mp + (size_t)SPLIT * HW;          // SPLIT*HW

    ssa_conv_kernel<<<C_ * (HW / 256), 256, 0, stream>>>(x, dW, db, cW, cb, y, v, yT, vT);
    ssa_flash_kernel<<<MBLK * SPLIT, WAVES * 32, 0, stream>>>(y, yT, vT, Op, mp, lp);
    ssa_combine_kernel<<<(C_ * HW) / 256, 256, 0, stream>>>(x, Op, mp, lp, (float*)d_out);
}